// LossHyperbolic_26654567039118
// MI455X (gfx1250) — compile-verified
//
#include <hip/hip_runtime.h>
#include <hip/hip_bf16.h>
#include <math.h>

typedef __attribute__((ext_vector_type(2))) float v2f;
typedef __attribute__((ext_vector_type(8))) float v8f;

#define BATCH 4096
#define DIM   256
#define NCLS  1000
#define CLIP_HI   0.99999f          // 1 - 1e-5
#define LOGQ_NEG  -11.512925464970229f  // log(1e-5)
#define XS_STRIDE 260               // 256 + pad (bank-conflict-free, 16B rows)
#define YS_STRIDE 68                // 64 + pad

// ---------------- prologue kernels ----------------

__global__ __launch_bounds__(256) void zero_hist_kernel(int* hist) {
    int t = blockIdx.x * 256 + threadIdx.x;
    if (t < 1024) hist[t] = 0;
}

__global__ __launch_bounds__(256) void hist_kernel(const int* __restrict__ labels, int* hist) {
    int i = blockIdx.x * 256 + threadIdx.x;
    if (i < BATCH) atomicAdd(&hist[labels[i]], 1);
}

// gather class embeddings + per-row scalars; one wave per row
__global__ __launch_bounds__(256) void prep_kernel(
    const float* __restrict__ img, const float* __restrict__ txt,
    const float* __restrict__ clsw, const int* __restrict__ labels,
    const int* __restrict__ hist,
    float* __restrict__ xbuf, float* __restrict__ xs2, float* __restrict__ xsn,
    float* __restrict__ psi, float* __restrict__ lqp,
    float* __restrict__ y2i, float* __restrict__ yni,
    float* __restrict__ y2t, float* __restrict__ ynt)
{
    int wave = threadIdx.x >> 5, lane = threadIdx.x & 31;
    int i = blockIdx.x * 8 + wave;
    int lab = labels[i];
    const float* crow = clsw + (size_t)lab * DIM;
    float sx = 0.f, si = 0.f, st = 0.f;
#pragma unroll
    for (int q = 0; q < 8; ++q) {
        int k = lane + 32 * q;
        float xv = crow[k];
        xbuf[(size_t)i * DIM + k] = xv;
        sx += xv * xv;
        float iv = img[(size_t)i * DIM + k]; si += iv * iv;
        float tv = txt[(size_t)i * DIM + k]; st += tv * tv;
    }
#pragma unroll
    for (int m = 16; m >= 1; m >>= 1) {
        sx += __shfl_xor(sx, m, 32);
        si += __shfl_xor(si, m, 32);
        st += __shfl_xor(st, m, 32);
    }
    if (lane == 0) {
        xs2[i] = sx;
        float n = sqrtf(sx);
        xsn[i] = n;
        float a = 0.1f * (1.f - sx) / n;                 // K = 0.1
        a = fminf(fmaxf(a, -CLIP_HI), CLIP_HI);
        psi[i] = asinf(a);
        int cnt = hist[lab];
        lqp[i] = logf(1.f / (float)cnt + 1e-5f);         // log(mask_norm*E + eps), pos case
        y2i[i] = si; yni[i] = sqrtf(si);
        y2t[i] = st; ynt[i] = sqrtf(st);
    }
}

// ---------------- fused GEMM + entailment + hypernym kernel ----------------
// grid: (128 row-blocks of 32 rows, 2 targets), block: 256 threads = 8 waves
// wave (wm,wn): wm = wave>>2 in {0,1} (16-row tile), wn = wave&3 (16-col tile)
__global__ __launch_bounds__(256) void fused_kernel(
    const float* __restrict__ xbuf,
    const float* __restrict__ y_img, const float* __restrict__ y_txt,
    const float* __restrict__ xs2, const float* __restrict__ xsn,
    const float* __restrict__ psi, const float* __restrict__ lqp,
    const float* __restrict__ y2i, const float* __restrict__ yni,
    const float* __restrict__ y2t, const float* __restrict__ ynt,
    const int* __restrict__ labels,
    float* __restrict__ ent_part, float* __restrict__ hyp_part)
{
    const int tgt = blockIdx.y;
    const float* __restrict__ y   = tgt ? y_txt : y_img;
    const float* __restrict__ y2a = tgt ? y2t : y2i;
    const float* __restrict__ yna = tgt ? ynt : yni;

    __shared__ float xs[32 * XS_STRIDE];     // 32 rows x 256 K  (padded)
    __shared__ float ys[64 * YS_STRIDE];     // 64 cols x 64 K chunk (padded)
    __shared__ float red[32][4][4];          // per-row, per-wn: m, l, t
    __shared__ float entw[8];

    const int tid  = threadIdx.x;
    const int wave = tid >> 5, lane = tid & 31;
    const int wm = wave >> 2, wn = wave & 3;
    const int h  = lane >> 4, ml = lane & 15;
    const int rb0 = blockIdx.x * 32;

    // stage A tile (32 x 256) into LDS, coalesced float4
#pragma unroll
    for (int q = 0; q < 8; ++q) {
        int idx = tid + 256 * q;            // 2048 float4
        int row = idx >> 6;                 // 64 float4 per row
        int c4  = idx & 63;
        const float4 v = *(const float4*)(xbuf + (size_t)(rb0 + row) * DIM + c4 * 4);
        *(float4*)(&xs[row * XS_STRIDE + c4 * 4]) = v;
    }

    // per-lane row scalars for the 8 C rows this lane owns (M = r + 8h)
    float x2r[8], xnr[8], psr[8], lqr[8], bcr[8];
    int   lbr[8];
#pragma unroll
    for (int r = 0; r < 8; ++r) {
        int i = rb0 + wm * 16 + r + 8 * h;
        x2r[r] = xs2[i]; xnr[r] = xsn[i]; psr[r] = psi[i];
        lqr[r] = lqp[i]; lbr[r] = labels[i];
        bcr[r] = 1.f - x2r[r];
    }

    // online-softmax state per owned row (s = -d)
    float mso[8], lso[8], tso[8];
#pragma unroll
    for (int r = 0; r < 8; ++r) { mso[r] = -3.0e38f; lso[r] = 0.f; tso[r] = 0.f; }
    float entacc = 0.f;

    for (int ct = 0; ct < 64; ++ct) {        // column tiles of 64
        const int colBase = ct * 64;
        v8f c = {};
        for (int kc = 0; kc < 4; ++kc) {     // K chunks of 64
            __syncthreads();                 // protect ys reuse (also covers xs on first pass)
#pragma unroll
            for (int q = 0; q < 4; ++q) {    // stage B chunk: 64 cols x 64 K
                int idx = tid + 256 * q;     // 1024 float4
                int row = idx >> 4;          // 16 float4 per row
                int c4  = idx & 15;
                const float4 v = *(const float4*)(y + (size_t)(colBase + row) * DIM + kc * 64 + c4 * 4);
                *(float4*)(&ys[row * YS_STRIDE + c4 * 4]) = v;
            }
            __syncthreads();
            // A frag: x[m][k0+2h, k0+2h+1]; B frag: y[n][k0+2h, k0+2h+1] (ISA 16x4 / 4x16 layouts)
            const float* xrow = &xs[(wm * 16 + ml) * XS_STRIDE + kc * 64 + 2 * h];
            const float* yrow = &ys[(wn * 16 + ml) * YS_STRIDE + 2 * h];
#pragma unroll
            for (int ks = 0; ks < 16; ++ks) {
                v2f a = *(const v2f*)(xrow + ks * 4);
                v2f b = *(const v2f*)(yrow + ks * 4);
                c = __builtin_amdgcn_wmma_f32_16x16x4_f32(
                        false, a, false, b, (short)0, c, false, false);
            }
        }
        // ---- fused tail on the 16x16 dot tile ----
        const int j = colBase + wn * 16 + ml;
        const float y2j = y2a[j];
        const float ynj = yna[j];
        const int   lj  = labels[j];
#pragma unroll
        for (int r = 0; r < 8; ++r) {
            const float dot = c[r];
            const float x2 = x2r[r], xn = xnr[r], bc = bcr[r];
            // entailment E operator
            float xy_dist = sqrtf(fmaxf(x2 + y2j - 2.f * dot, 0.f));
            float inner   = fmaxf(1.f + x2 * y2j - 2.f * dot, 0.f);   // 1+(xn*yn)^2-2dot
            float den     = xn * xy_dist * sqrtf(inner);
            float arg     = (dot * (1.f + x2) - x2 * (1.f + y2j)) / den;
            arg = fminf(fmaxf(arg, -CLIP_HI), CLIP_HI);
            float theta = acosf(arg);
            float E = fmaxf(theta - psr[r], 0.f);
            bool pos = (lbr[r] == lj);
            entacc += pos ? E : fmaxf(1.0f - E, 0.f);                 // ALPHA = 1
            // hyperbolic distance (c=1): xy = -dot
            float A  = 1.f - 2.f * dot + y2j;
            float ns = A * A * x2 - 2.f * A * bc * dot + bc * bc * y2j;
            float dd = (1.f - 2.f * dot + x2 * y2j) + 1e-5f;
            float nc = fminf(sqrtf(fmaxf(ns, 0.f)) / dd, CLIP_HI);
            float dv = logf((1.f + nc) / (1.f - nc));                 // 2*artanh(nc)
            float s  = -dv;
            float logq = pos ? lqr[r] : LOGQ_NEG;
            // online softmax update
            float mn = fmaxf(mso[r], s);
            float sc = __expf(mso[r] - mn);
            float es = __expf(s - mn);
            lso[r] = lso[r] * sc + es;
            tso[r] = tso[r] * sc + es * (s - logq);
            mso[r] = mn;
        }
    }

    // merge softmax state across the 16 lanes of each half-wave (same rows)
#pragma unroll
    for (int r = 0; r < 8; ++r) {
        float m = mso[r], l = lso[r], t = tso[r];
#pragma unroll
        for (int msk = 1; msk <= 8; msk <<= 1) {
            float m2 = __shfl_xor(m, msk, 32);
            float l2 = __shfl_xor(l, msk, 32);
            float t2 = __shfl_xor(t, msk, 32);
            float mn = fmaxf(m, m2);
            float s1 = __expf(m - mn), s2 = __expf(m2 - mn);
            l = l * s1 + l2 * s2;
            t = t * s1 + t2 * s2;
            m = mn;
        }
        mso[r] = m; lso[r] = l; tso[r] = t;
    }
    if (ml == 0) {
#pragma unroll
        for (int r = 0; r < 8; ++r) {
            int row_local = wm * 16 + r + 8 * h;
            red[row_local][wn][0] = mso[r];
            red[row_local][wn][1] = lso[r];
            red[row_local][wn][2] = tso[r];
        }
    }
    // entailment wave reduction
    float e = entacc;
#pragma unroll
    for (int msk = 16; msk >= 1; msk >>= 1) e += __shfl_xor(e, msk, 32);
    if (lane == 0) entw[wave] = e;
    __syncthreads();

    if (tid < 32) {
        float m = red[tid][0][0], l = red[tid][0][1], t = red[tid][0][2];
#pragma unroll
        for (int w = 1; w < 4; ++w) {
            float m2 = red[tid][w][0], l2 = red[tid][w][1], t2 = red[tid][w][2];
            float mn = fmaxf(m, m2);
            float s1 = __expf(m - mn), s2 = __expf(m2 - mn);
            l = l * s1 + l2 * s2;
            t = t * s1 + t2 * s2;
            m = mn;
        }
        float loss = t / l - (m + logf(l));   // sum_j p*(logp - logq) for this row
#pragma unroll
        for (int msk = 16; msk >= 1; msk >>= 1) loss += __shfl_xor(loss, msk, 32);
        if (tid == 0) {
            hyp_part[blockIdx.y * 128 + blockIdx.x] = loss;
            float es = 0.f;
#pragma unroll
            for (int w = 0; w < 8; ++w) es += entw[w];
            ent_part[blockIdx.y * 128 + blockIdx.x] = es;
        }
    }
}

// ---------------- final reduction ----------------
__global__ __launch_bounds__(256) void final_kernel(
    const float* __restrict__ ent_part, const float* __restrict__ hyp_part,
    float* __restrict__ out)
{
    __shared__ float se[8], sh[8];
    int tid = threadIdx.x, wave = tid >> 5, lane = tid & 31;
    float e = ent_part[tid];   // 2 targets x 128 blocks = 256 entries
    float hh = hyp_part[tid];
#pragma unroll
    for (int msk = 16; msk >= 1; msk >>= 1) {
        e  += __shfl_xor(e, msk, 32);
        hh += __shfl_xor(hh, msk, 32);
    }
    if (lane == 0) { se[wave] = e; sh[wave] = hh; }
    __syncthreads();
    if (tid == 0) {
        float E = 0.f, H = 0.f;
#pragma unroll
        for (int w = 0; w < 8; ++w) { E += se[w]; H += sh[w]; }
        H /= (float)BATCH;                 // jnp.mean over rows (both targets summed)
        out[0] = E;
        out[1] = H;
        out[2] = E + H;
    }
}

// ---------------- launch ----------------
extern "C" void kernel_launch(void* const* d_in, const int* in_sizes, int n_in,
                              void* d_out, int out_size, void* d_ws, size_t ws_size,
                              hipStream_t stream) {
    const float* img   = (const float*)d_in[0];
    const float* txt   = (const float*)d_in[1];
    const float* clsw  = (const float*)d_in[2];
    const int*   labels= (const int*)d_in[3];
    float* out = (float*)d_out;

    float* ws = (float*)d_ws;
    float* xbuf = ws;                          // 4096*256
    float* xs2  = xbuf + (size_t)BATCH * DIM;  // 4096 each below
    float* xsn  = xs2 + BATCH;
    float* psi  = xsn + BATCH;
    float* lqp  = psi + BATCH;
    float* y2i  = lqp + BATCH;
    float* yni  = y2i + BATCH;
    float* y2t  = yni + BATCH;
    float* ynt  = y2t + BATCH;
    float* ent_part = ynt + BATCH;             // 256
    float* hyp_part = ent_part + 256;          // 256
    int*   hist = (int*)(hyp_part + 256);      // 1024 ints

    zero_hist_kernel<<<4, 256, 0, stream>>>(hist);
    hist_kernel<<<BATCH / 256, 256, 0, stream>>>(labels, hist);
    prep_kernel<<<BATCH / 8, 256, 0, stream>>>(img, txt, clsw, labels, hist,
                                               xbuf, xs2, xsn, psi, lqp,
                                               y2i, yni, y2t, ynt);
    dim3 grid(BATCH / 32, 2);
    fused_kernel<<<grid, 256, 0, stream>>>(xbuf, img, txt, xs2, xsn, psi, lqp,
                                           y2i, yni, y2t, ynt, labels,
                                           ent_part, hyp_part);
    final_kernel<<<1, 256, 0, stream>>>(ent_part, hyp_part, out);
}